// Model_68530498175322
// MI455X (gfx1250) — compile-verified
//
#include <hip/hip_runtime.h>

// Problem constants (from reference setup_inputs)
#define B_  4096
#define S_  50
#define M_  20
#define D_  512
#define BN_EPS 1e-5f

// LDS rows: data rows + ONE shared zero row for WMMA tile padding.
#define IMG_ROWS (S_ + 1)   // row 50 = zeros
#define TXT_ROWS (M_ + 1)   // row 20 = zeros
#define AW_COLS  64         // 4 s-tiles of 16

typedef __attribute__((ext_vector_type(2))) float v2f;
typedef __attribute__((ext_vector_type(8))) float v8f;
typedef int b128i __attribute__((vector_size(4 * sizeof(int))));  // matches builtin param

// Dynamic LDS partition (floats)
#define SMEM_FLOATS (IMG_ROWS*D_ + TXT_ROWS*D_ + M_*AW_COLS + 64 + 32 + D_ + D_ + 256 + 8)
#define SMEM_BYTES  (SMEM_FLOATS * 4)   // ~155 KB -> 2 workgroups per 320KB WGP

#define GLOBAL_AS __attribute__((address_space(1)))
#define LDS_AS    __attribute__((address_space(3)))

#if defined(__AMDGCN__) && __has_builtin(__builtin_amdgcn_global_load_async_to_lds_b128) && __has_builtin(__builtin_amdgcn_s_wait_asynccnt)
#define USE_ASYNC 1
#else
#define USE_ASYNC 0
#endif

__device__ __forceinline__ float wave_reduce_sum(float v) {
    // wave32 butterfly reduction
    #pragma unroll
    for (int off = 16; off > 0; off >>= 1) v += __shfl_xor(v, off, 32);
    return v;
}

__global__ void __launch_bounds__(256)
fused_attnpool_kernel(const float* __restrict__ img_g,   // [B,S,D]
                      const float* __restrict__ txt_g,   // [M,2,D]
                      const int*   __restrict__ rad_g,   // [B,M]
                      const int*   __restrict__ msk_g,   // [B,S]
                      const float* __restrict__ gamma,   // [2D]
                      const float* __restrict__ beta,    // [2D]
                      const float* __restrict__ bmean,   // [2D]
                      const float* __restrict__ bvar,    // [2D]
                      const float* __restrict__ wcls,    // [2,2D]
                      const float* __restrict__ bcls,    // [2]
                      float*       __restrict__ out)     // [B,2]
{
    extern __shared__ float smem[];
    float* img = smem;                     // IMG_ROWS*D  (row 50 = zeros)
    float* txt = img + IMG_ROWS * D_;      // TXT_ROWS*D  (row 20 = zeros)
    float* aw  = txt + TXT_ROWS * D_;      // M_*AW_COLS
    float* ai  = aw  + M_ * AW_COLS;       // 64
    float* at  = ai  + 64;                 // 32
    float* ip  = at  + 32;                 // D
    float* tp  = ip  + D_;                 // D
    float* red = tp  + D_;                 // 256
    float* sc  = red + 256;                // [0]=cnt, [1]=1/|ip|, [2]=1/|tp|

    const int b    = blockIdx.x;
    const int tid  = threadIdx.x;
    const int wave = tid >> 5;
    const int lane = tid & 31;

#if USE_ASYNC
    // ---- Fire-and-forget async copy of raw image_feats[b] into LDS ----
    // 50*512 floats = 6400 x b128 lane-ops; tracked by ASYNCcnt, no VGPR staging.
    {
        const float* base = img_g + (size_t)b * S_ * D_;
        for (int c = tid; c < (S_ * D_) / 4; c += 256) {
            __builtin_amdgcn_global_load_async_to_lds_b128(
                (GLOBAL_AS b128i*)(base + 4 * c),
                (LDS_AS b128i*)(img + 4 * c), 0, 0);
        }
    }
#endif

    // ---- shared zero rows for WMMA tile padding ----
    for (int i = tid; i < D_; i += 256) {
        img[S_ * D_ + i] = 0.0f;
        txt[M_ * D_ + i] = 0.0f;
    }

    if (tid == 0) {
        int c = 0;
        for (int s = 0; s < S_; ++s) c += (msk_g[b * S_ + s] == 1);
        sc[0] = fmaxf((float)c, 1.0f);
    }

    // ---- text: ragged gather (L2-resident, 80KB total) + L2-norm into LDS ----
    // Overlaps with the in-flight async image copy.
    for (int m = wave; m < M_; m += 8) {
        const int rsel = rad_g[b * M_ + m];
        const float* row = txt_g + (size_t)(m * 2 + rsel) * D_;
        float x[16];
        float ss = 0.0f;
        #pragma unroll
        for (int i = 0; i < 16; ++i) { x[i] = row[lane + 32 * i]; ss += x[i] * x[i]; }
        ss = wave_reduce_sum(ss);
        const float r = rsqrtf(ss);
        #pragma unroll
        for (int i = 0; i < 16; ++i) txt[m * D_ + lane + 32 * i] = x[i] * r;
    }

#if USE_ASYNC
    __builtin_amdgcn_s_wait_asynccnt(0);   // my async writes landed in LDS
    __syncthreads();                       // everyone's landed
    // ---- L2-normalize + mask image rows in place in LDS ----
    for (int s = wave; s < S_; s += 8) {
        float x[16];
        float ss = 0.0f;
        #pragma unroll
        for (int i = 0; i < 16; ++i) { x[i] = img[s * D_ + lane + 32 * i]; ss += x[i] * x[i]; }
        ss = wave_reduce_sum(ss);
        const float r = (msk_g[b * S_ + s] == 1) ? rsqrtf(ss) : 0.0f;
        #pragma unroll
        for (int i = 0; i < 16; ++i) img[s * D_ + lane + 32 * i] = x[i] * r;
    }
#else
    // ---- fallback: register-staged load + L2-norm + mask into LDS ----
    for (int s = wave; s < S_; s += 8) {
        const float* row = img_g + ((size_t)b * S_ + s) * D_;
        float x[16];
        float ss = 0.0f;
        #pragma unroll
        for (int i = 0; i < 16; ++i) { x[i] = row[lane + 32 * i]; ss += x[i] * x[i]; }
        ss = wave_reduce_sum(ss);
        const float r = (msk_g[b * S_ + s] == 1) ? rsqrtf(ss) : 0.0f;
        #pragma unroll
        for (int i = 0; i < 16; ++i) img[s * D_ + lane + 32 * i] = x[i] * r;
    }
#endif

    __syncthreads();

    // ---- aw[20 x 50] = txt(M x D) * img^T via V_WMMA_F32_16X16X4_F32 ----
    // 8 waves -> 8 (mt,st) 16x16 tiles; 128 K-steps of K=4.
    // Out-of-range rows clamp to the shared zero row (contributes 0).
    {
        const int mt   = wave >> 2;      // 0..1
        const int st   = wave & 3;       // 0..3
        const int half = lane >> 4;      // selects K pair {0,1} vs {2,3}
        const int l16  = lane & 15;
        const int mrow = mt * 16 + l16;
        const int srow = st * 16 + l16;
        const float* arow = txt + min(mrow, M_) * D_;   // A: row M (zero row if padded)
        const float* brow = img + min(srow, S_) * D_;   // B: col N of img^T
        v8f acc = {0.f, 0.f, 0.f, 0.f, 0.f, 0.f, 0.f, 0.f};
        #pragma unroll 8
        for (int k = 0; k < D_; k += 4) {
            const int ko = k + 2 * half;
            v2f a, bb;
            a.x  = arow[ko]; a.y  = arow[ko + 1];
            bb.x = brow[ko]; bb.y = brow[ko + 1];
            acc = __builtin_amdgcn_wmma_f32_16x16x4_f32(
                /*neg_a=*/false, a, /*neg_b=*/false, bb,
                /*c_mod=*/(short)0, acc, /*reuse_a=*/false, /*reuse_b=*/false);
        }
        // C/D layout: VGPR v, lanes 0-15 -> M=v, lanes 16-31 -> M=v+8, N=lane&15
        #pragma unroll
        for (int v = 0; v < 8; ++v) {
            const int rowi = mt * 16 + v + 8 * half;
            const int coli = st * 16 + l16;
            if (rowi < M_) aw[rowi * AW_COLS + coli] = acc[v];
        }
    }
    __syncthreads();

    // ---- attn_image (mean over M) and attn_text (sum over S / cnt) ----
    if (tid < S_) {
        float s = 0.0f;
        for (int m = 0; m < M_; ++m) s += aw[m * AW_COLS + tid];
        ai[tid] = s * (1.0f / (float)M_);
    }
    if (tid >= 64 && tid < 64 + M_) {
        const int m = tid - 64;
        float s = 0.0f;
        for (int c = 0; c < S_; ++c) s += aw[m * AW_COLS + c];
        at[m] = s / sc[0];
    }
    __syncthreads();

    // ---- pooled vectors from LDS-resident img/txt ----
    for (int d = tid; d < D_; d += 256) {
        float s = 0.0f;
        for (int x = 0; x < S_; ++x) s += ai[x] * img[x * D_ + d];
        ip[d] = s;
        float t = 0.0f;
        for (int m = 0; m < M_; ++m) t += at[m] * txt[m * D_ + d];
        tp[d] = t;
    }
    __syncthreads();

    // ---- L2 norms of both pooled vectors (block reduction) ----
    {
        float pi = 0.0f, pt = 0.0f;
        for (int d = tid; d < D_; d += 256) { pi += ip[d] * ip[d]; pt += tp[d] * tp[d]; }
        pi = wave_reduce_sum(pi);
        pt = wave_reduce_sum(pt);
        if (lane == 0) { red[wave] = pi; red[8 + wave] = pt; }
    }
    __syncthreads();
    if (tid == 0) {
        float a = 0.0f, c = 0.0f;
        for (int w = 0; w < 8; ++w) { a += red[w]; c += red[8 + w]; }
        sc[1] = rsqrtf(a);
        sc[2] = rsqrtf(c);
    }
    __syncthreads();

    // ---- ReLU -> BatchNorm(eval) -> 1024x2 classifier, fused block reduction ----
    {
        float d0 = 0.0f, d1 = 0.0f;
        for (int d = tid; d < 2 * D_; d += 256) {
            float v = (d < D_) ? ip[d] * sc[1] : tp[d - D_] * sc[2];
            v = fmaxf(v, 0.0f);
            v = (v - bmean[d]) * rsqrtf(bvar[d] + BN_EPS) * gamma[d] + beta[d];
            d0 += v * wcls[d];
            d1 += v * wcls[2 * D_ + d];
        }
        d0 = wave_reduce_sum(d0);
        d1 = wave_reduce_sum(d1);
        if (lane == 0) { red[wave] = d0; red[8 + wave] = d1; }
    }
    __syncthreads();
    if (tid == 0) {
        float a = 0.0f, c = 0.0f;
        for (int w = 0; w < 8; ++w) { a += red[w]; c += red[8 + w]; }
        out[b * 2 + 0] = a + bcls[0];
        out[b * 2 + 1] = c + bcls[1];
    }
}

extern "C" void kernel_launch(void* const* d_in, const int* in_sizes, int n_in,
                              void* d_out, int out_size, void* d_ws, size_t ws_size,
                              hipStream_t stream) {
    const float* img   = (const float*)d_in[0];
    const float* txt   = (const float*)d_in[1];
    const int*   rad   = (const int*)  d_in[2];
    const int*   msk   = (const int*)  d_in[3];
    const float* gamma = (const float*)d_in[4];
    const float* beta  = (const float*)d_in[5];
    const float* bmean = (const float*)d_in[6];
    const float* bvar  = (const float*)d_in[7];
    const float* wcls  = (const float*)d_in[8];
    const float* bcls  = (const float*)d_in[9];
    float* out = (float*)d_out;

    // ~155 KB dynamic LDS per workgroup -> 2 concurrent WGs per 320KB WGP.
    // hipFuncSetAttribute is not a stream op; capture-safe and deterministic.
    (void)hipFuncSetAttribute((const void*)fused_attnpool_kernel,
                              hipFuncAttributeMaxDynamicSharedMemorySize,
                              SMEM_BYTES);

    fused_attnpool_kernel<<<B_, 256, SMEM_BYTES, stream>>>(
        img, txt, rad, msk, gamma, beta, bmean, bvar, wcls, bcls, out);
}